// model2_3l_30073361006597
// MI455X (gfx1250) — compile-verified
//
#include <hip/hip_runtime.h>
#include <cstdint>

typedef __attribute__((ext_vector_type(2))) float v2f;
typedef __attribute__((ext_vector_type(8))) float v8f;

#define BN_EPS 1e-5f

// ---------------- degree ----------------
__global__ void k_deg(const int* __restrict__ ei, float* __restrict__ deg, int n_edges) {
  int e = blockIdx.x * blockDim.x + threadIdx.x;
  if (e < n_edges) atomicAdd(&deg[ei[n_edges + e]], 1.0f);
}
__global__ void k_inv_deg(float* __restrict__ deg, int n) {
  int i = blockIdx.x * blockDim.x + threadIdx.x;
  if (i < n) deg[i] = 1.0f / fmaxf(deg[i], 1.0f);
}

// ---------------- layer-1 input padding (3 -> 4 channels, W1 -> 4x4x8) ----------------
__global__ void k_pad_x(const float* __restrict__ x, float* __restrict__ xp, int n) {
  int i = blockIdx.x * blockDim.x + threadIdx.x;
  if (i < n) {
    xp[4 * i + 0] = x[3 * i + 0];
    xp[4 * i + 1] = x[3 * i + 1];
    xp[4 * i + 2] = x[3 * i + 2];
    xp[4 * i + 3] = 0.0f;
  }
}
__global__ void k_pad_w1(const float* __restrict__ W1, float* __restrict__ Wp) {
  int i = threadIdx.x;  // 4*4*8 = 128 threads
  if (i < 128) {
    int k = i >> 5;
    int r = i & 31;
    int c = r >> 3;
    int d = r & 7;
    Wp[i] = (c < 3) ? W1[(k * 3 + c) * 8 + d] : 0.0f;
  }
}

// ---------------- node transform: T[n, k*COUT+d] = sum_c h[n,c] * W[k,c,d] ----------------
// Full 16-row tiles only; CIN must be a multiple of 4 -> zero divergence, EXEC all-1s.
// One wave computes one 16x16 output tile via V_WMMA_F32_16X16X4_F32.
template <int CIN, int NCOLS>
__global__ void k_node_transform(const float* __restrict__ h,
                                 const float* __restrict__ W,   // [4][CIN][NCOLS/4]
                                 float* __restrict__ T,
                                 int mTiles) {
  static_assert(CIN % 4 == 0, "CIN must be multiple of 4");
  const int lane = threadIdx.x & 31;
  const int wave = threadIdx.x >> 5;
  const int wavesPerBlock = blockDim.x >> 5;
  const int tile = blockIdx.x * wavesPerBlock + wave;

  const int nTilesN = NCOLS / 16;
  if (tile >= mTiles * nTilesN) return;   // uniform per wave

  const int mTile = tile / nTilesN;
  const int nTile = tile % nTilesN;
  const int COUT = NCOLS / 4;

  const int rowA = mTile * 16 + (lane & 15);     // A: M index for this lane
  const int col  = nTile * 16 + (lane & 15);     // B/D: N index for this lane
  const int khalf = (lane >> 4) * 2;             // lanes 16-31 hold upper K pair

  const int kk = col / COUT;                     // spline weight index
  const int dd = col - kk * COUT;                // output channel

  const float* __restrict__ hrow = h + (size_t)rowA * CIN;
  const float* __restrict__ wcol = W + (size_t)kk * CIN * COUT + dd;

  v8f acc = {};
#pragma unroll
  for (int k0 = 0; k0 < CIN; k0 += 4) {
    const int c0 = k0 + khalf;
    v2f a, b;
    a.x = hrow[c0];
    a.y = hrow[c0 + 1];
    b.x = wcol[(size_t)(c0)     * COUT];
    b.y = wcol[(size_t)(c0 + 1) * COUT];
    acc = __builtin_amdgcn_wmma_f32_16x16x4_f32(false, a, false, b, (short)0, acc, false, false);
  }

  float* __restrict__ tout = T + ((size_t)mTile * 16 + (lane >> 4) * 8) * NCOLS + col;
#pragma unroll
  for (int r = 0; r < 8; ++r) tout[(size_t)r * NCOLS] = acc[r];
}

// scalar tail for n_nodes % 16 rows (generic correctness; not launched when divisible)
template <int CIN, int NCOLS>
__global__ void k_node_tail(const float* __restrict__ h, const float* __restrict__ W,
                            float* __restrict__ T, int rowStart, int n_nodes) {
  const int COUT = NCOLS / 4;
  int idx = blockIdx.x * blockDim.x + threadIdx.x;
  int row = rowStart + idx / NCOLS;
  int col = idx % NCOLS;
  if (row >= n_nodes) return;
  int kk = col / COUT, dd = col - kk * COUT;
  float s = 0.0f;
  for (int c = 0; c < CIN; ++c) s += h[(size_t)row * CIN + c] * W[(kk * CIN + c) * COUT + dd];
  T[(size_t)row * NCOLS + col] = s;
}

// ---------------- edge gather + scatter-add (float4 gathers) ----------------
template <int COUT>
__global__ void k_edge_scatter(const int* __restrict__ ei, const float* __restrict__ ea,
                               const float* __restrict__ T, float* __restrict__ agg,
                               int n_edges) {
  int e = blockIdx.x * blockDim.x + threadIdx.x;
  if (e >= n_edges) return;
  const int src = ei[e];
  const int dst = ei[n_edges + e];
  const float2 u = ((const float2*)ea)[e];
  const float u0 = fminf(fmaxf(u.x, 0.0f), 1.0f);
  const float u1 = fminf(fmaxf(u.y, 0.0f), 1.0f);
  const float b0 = (1.0f - u0) * (1.0f - u1);
  const float b1 = u0 * (1.0f - u1);
  const float b2 = (1.0f - u0) * u1;
  const float b3 = u0 * u1;
  const float4* __restrict__ t4 = (const float4*)(T + (size_t)src * (4 * COUT));
  float* __restrict__ a = agg + (size_t)dst * COUT;
  constexpr int Q = COUT / 4;
#pragma unroll
  for (int q = 0; q < Q; ++q) {
    const float4 t0 = t4[q];
    const float4 t1 = t4[Q + q];
    const float4 t2 = t4[2 * Q + q];
    const float4 t3 = t4[3 * Q + q];
    atomicAdd(&a[4 * q + 0], b0 * t0.x + b1 * t1.x + b2 * t2.x + b3 * t3.x);
    atomicAdd(&a[4 * q + 1], b0 * t0.y + b1 * t1.y + b2 * t2.y + b3 * t3.y);
    atomicAdd(&a[4 * q + 2], b0 * t0.z + b1 * t1.z + b2 * t2.z + b3 * t3.z);
    atomicAdd(&a[4 * q + 3], b0 * t0.w + b1 * t1.w + b2 * t2.w + b3 * t3.w);
  }
}

// ---------------- scale by inv_deg, ELU, accumulate BN stats (LDS reduction) ----------------
template <int COUT>
__global__ void k_elu_stats(const float* __restrict__ agg, const float* __restrict__ invdeg,
                            float* __restrict__ h, float* __restrict__ stats /*2*COUT*/,
                            int n) {
  __shared__ float ssum[COUT];
  __shared__ float ssq[COUT];
  for (int d = threadIdx.x; d < COUT; d += blockDim.x) { ssum[d] = 0.0f; ssq[d] = 0.0f; }
  __syncthreads();
  const int i = blockIdx.x * blockDim.x + threadIdx.x;
  if (i < n) {
    const float s = invdeg[i];
#pragma unroll
    for (int d = 0; d < COUT; ++d) {
      float v = agg[(size_t)i * COUT + d] * s;
      v = (v > 0.0f) ? v : (expf(v) - 1.0f);
      h[(size_t)i * COUT + d] = v;
      atomicAdd(&ssum[d], v);        // ds_add_f32
      atomicAdd(&ssq[d], v * v);
    }
  }
  __syncthreads();
  for (int d = threadIdx.x; d < COUT; d += blockDim.x) {
    atomicAdd(&stats[d], ssum[d]);
    atomicAdd(&stats[COUT + d], ssq[d]);
  }
}

// ---------------- batchnorm apply ----------------
template <int COUT>
__global__ void k_bn_apply(float* __restrict__ h, const float* __restrict__ stats,
                           const float* __restrict__ g, const float* __restrict__ b, int n) {
  const int i = blockIdx.x * blockDim.x + threadIdx.x;
  if (i >= n) return;
  const float inv_n = 1.0f / (float)n;
#pragma unroll
  for (int d = 0; d < COUT; ++d) {
    const float mu = stats[d] * inv_n;
    const float var = stats[COUT + d] * inv_n - mu * mu;
    const float rs = rsqrtf(var + BN_EPS);
    h[(size_t)i * COUT + d] = (h[(size_t)i * COUT + d] - mu) * rs * g[d] + b[d];
  }
}

// ---------------- segment-mean pooling (LDS-staged) ----------------
__global__ void k_pool(const float* __restrict__ h, const int* __restrict__ batch,
                       float* __restrict__ pooled /*64*32*/, float* __restrict__ cnt /*64*/,
                       int n) {
  __shared__ float sp[64 * 32];
  __shared__ float sc[64];
  for (int i = threadIdx.x; i < 64 * 32; i += blockDim.x) sp[i] = 0.0f;
  for (int i = threadIdx.x; i < 64; i += blockDim.x) sc[i] = 0.0f;
  __syncthreads();
  const int i = blockIdx.x * blockDim.x + threadIdx.x;
  if (i < n) {
    const int g = batch[i];
    atomicAdd(&sc[g], 1.0f);
#pragma unroll
    for (int d = 0; d < 32; ++d) atomicAdd(&sp[g * 32 + d], h[(size_t)i * 32 + d]);
  }
  __syncthreads();
  for (int j = threadIdx.x; j < 64 * 32; j += blockDim.x)
    if (sp[j] != 0.0f) atomicAdd(&pooled[j], sp[j]);
  for (int j = threadIdx.x; j < 64; j += blockDim.x)
    if (sc[j] != 0.0f) atomicAdd(&cnt[j], sc[j]);
}

// ---------------- final FC: out[g,j] = (pooled[g,:]/cnt[g]) . fc_w[j,:] ----------------
__global__ void k_fc(const float* __restrict__ pooled, const float* __restrict__ cnt,
                     const float* __restrict__ fcw, float* __restrict__ out) {
  const int idx = threadIdx.x;
  if (idx >= 64 * 10) return;
  const int g = idx / 10;
  const int j = idx - g * 10;
  const float ic = 1.0f / fmaxf(cnt[g], 1.0f);
  float s = 0.0f;
#pragma unroll
  for (int d = 0; d < 32; ++d) s += pooled[g * 32 + d] * ic * fcw[j * 32 + d];
  out[idx] = s;
}

static inline int cdiv(long a, long b) { return (int)((a + b - 1) / b); }

extern "C" void kernel_launch(void* const* d_in, const int* in_sizes, int n_in,
                              void* d_out, int out_size, void* d_ws, size_t ws_size,
                              hipStream_t stream) {
  const float* x     = (const float*)d_in[0];
  const int*   ei    = (const int*)  d_in[1];
  const float* ea    = (const float*)d_in[2];
  const int*   batch = (const int*)  d_in[3];
  const float* W1    = (const float*)d_in[4];
  const float* W2    = (const float*)d_in[5];
  const float* W3    = (const float*)d_in[6];
  const float* g1    = (const float*)d_in[7];
  const float* b1    = (const float*)d_in[8];
  const float* g2    = (const float*)d_in[9];
  const float* b2    = (const float*)d_in[10];
  const float* g3    = (const float*)d_in[11];
  const float* b3    = (const float*)d_in[12];
  const float* fcw   = (const float*)d_in[13];
  float* out = (float*)d_out;

  const int n_nodes = in_sizes[0] / 3;
  const int n_edges = in_sizes[1] / 2;

  // ---- workspace layout (floats) ----
  float* ws = (float*)d_ws;
  size_t off = 0;
  auto walloc = [&](size_t nf) { float* p = ws + off; off += (nf + 63) & ~(size_t)63; return p; };
  float* invdeg = walloc((size_t)n_nodes);
  float* xp     = walloc((size_t)n_nodes * 4);   // padded layer-1 input
  float* Wp     = walloc(4 * 4 * 8);             // padded W1
  float* T      = walloc((size_t)n_nodes * 128);
  float* agg    = walloc((size_t)n_nodes * 32);
  float* hA     = walloc((size_t)n_nodes * 32);
  float* hB     = walloc((size_t)n_nodes * 32);
  float* stats  = walloc(64);
  float* pooled = walloc(64 * 32);
  float* cnt    = walloc(64);
  (void)ws_size;

  const int TPB = 256;
  const int edgeBlocks = cdiv(n_edges, TPB);
  const int nodeBlocks = cdiv(n_nodes, TPB);
  const int mTilesFull = n_nodes / 16;
  const int tailRows = n_nodes - mTilesFull * 16;

  // ---- degrees + layer-1 padding ----
  hipMemsetAsync(invdeg, 0, (size_t)n_nodes * sizeof(float), stream);
  k_deg<<<edgeBlocks, TPB, 0, stream>>>(ei, invdeg, n_edges);
  k_inv_deg<<<nodeBlocks, TPB, 0, stream>>>(invdeg, n_nodes);
  k_pad_x<<<nodeBlocks, TPB, 0, stream>>>(x, xp, n_nodes);
  k_pad_w1<<<1, 128, 0, stream>>>(W1, Wp);

  // ---- layer 1: xp(N,4) -> hA(N,8) ----
  {
    const int NC = 32;
    k_node_transform<4, 32><<<cdiv((long)mTilesFull * (NC / 16), 8), TPB, 0, stream>>>(xp, Wp, T, mTilesFull);
    if (tailRows)
      k_node_tail<4, 32><<<cdiv((long)tailRows * NC, TPB), TPB, 0, stream>>>(xp, Wp, T, mTilesFull * 16, n_nodes);
    hipMemsetAsync(agg, 0, (size_t)n_nodes * 8 * sizeof(float), stream);
    hipMemsetAsync(stats, 0, 2 * 8 * sizeof(float), stream);
    k_edge_scatter<8><<<edgeBlocks, TPB, 0, stream>>>(ei, ea, T, agg, n_edges);
    k_elu_stats<8><<<nodeBlocks, TPB, 0, stream>>>(agg, invdeg, hA, stats, n_nodes);
    k_bn_apply<8><<<nodeBlocks, TPB, 0, stream>>>(hA, stats, g1, b1, n_nodes);
  }

  // ---- layer 2: hA(N,8) -> hB(N,16) ----
  {
    const int NC = 64;
    k_node_transform<8, 64><<<cdiv((long)mTilesFull * (NC / 16), 8), TPB, 0, stream>>>(hA, W2, T, mTilesFull);
    if (tailRows)
      k_node_tail<8, 64><<<cdiv((long)tailRows * NC, TPB), TPB, 0, stream>>>(hA, W2, T, mTilesFull * 16, n_nodes);
    hipMemsetAsync(agg, 0, (size_t)n_nodes * 16 * sizeof(float), stream);
    hipMemsetAsync(stats, 0, 2 * 16 * sizeof(float), stream);
    k_edge_scatter<16><<<edgeBlocks, TPB, 0, stream>>>(ei, ea, T, agg, n_edges);
    k_elu_stats<16><<<nodeBlocks, TPB, 0, stream>>>(agg, invdeg, hB, stats, n_nodes);
    k_bn_apply<16><<<nodeBlocks, TPB, 0, stream>>>(hB, stats, g2, b2, n_nodes);
  }

  // ---- layer 3: hB(N,16) -> hA(N,32) ----
  {
    const int NC = 128;
    k_node_transform<16, 128><<<cdiv((long)mTilesFull * (NC / 16), 8), TPB, 0, stream>>>(hB, W3, T, mTilesFull);
    if (tailRows)
      k_node_tail<16, 128><<<cdiv((long)tailRows * NC, TPB), TPB, 0, stream>>>(hB, W3, T, mTilesFull * 16, n_nodes);
    hipMemsetAsync(agg, 0, (size_t)n_nodes * 32 * sizeof(float), stream);
    hipMemsetAsync(stats, 0, 2 * 32 * sizeof(float), stream);
    k_edge_scatter<32><<<edgeBlocks, TPB, 0, stream>>>(ei, ea, T, agg, n_edges);
    k_elu_stats<32><<<nodeBlocks, TPB, 0, stream>>>(agg, invdeg, hA, stats, n_nodes);
    k_bn_apply<32><<<nodeBlocks, TPB, 0, stream>>>(hA, stats, g3, b3, n_nodes);
  }

  // ---- pooling + FC ----
  hipMemsetAsync(pooled, 0, 64 * 32 * sizeof(float), stream);
  hipMemsetAsync(cnt, 0, 64 * sizeof(float), stream);
  k_pool<<<nodeBlocks, TPB, 0, stream>>>(hA, batch, pooled, cnt, n_nodes);
  k_fc<<<1, 640, 0, stream>>>(pooled, cnt, fcw, out);
}